// FEMHeatSolver_43937515438339
// MI455X (gfx1250) — compile-verified
//
#include <hip/hip_runtime.h>
#include <hip/hip_bf16.h>
#include <stdint.h>

// FEM heat solver, MI455X (gfx1250, wave32).
// Bandwidth-bound: T/lap/Qt (77MB) stay resident in the 192MB L2 across all 13
// steps; the 333MB step history streams out with non-temporal b128 stores; the
// final transpose stages tiles via CDNA5 async global->LDS (ASYNCcnt) copies
// and emits fully coalesced output. SpMV scatter uses hardware
// global_atomic_add_f32 (no-return, STOREcnt) — never a CAS loop.

#define DT_CONST 0.01f
#define LANES 32          // BATCH == wave32 size
#define PLANE 1028        // 32*32 floats per t-plane in LDS + 4-float pad

typedef float v4f __attribute__((ext_vector_type(4)));

// ---------------------------------------------------------------------------
// Init: Qt[n*32+b] = x[b*N+n] (tiled 32x32 LDS transpose), T = lap = 0.
// ---------------------------------------------------------------------------
__global__ __launch_bounds__(256) void fem_init_kernel(
    const float* __restrict__ x, float* __restrict__ Qt,
    float* __restrict__ T, float* __restrict__ lap, int N) {
  __shared__ float tile[32 * 33];
  const int n0   = blockIdx.x * 32;
  const int lane = threadIdx.x & 31;
  const int w    = threadIdx.x >> 5;  // 0..7
#pragma unroll
  for (int k = 0; k < 4; ++k) {
    const int b = k * 8 + w;
    const int n = n0 + lane;
    tile[b * 33 + lane] = (n < N) ? x[(size_t)b * N + n] : 0.0f;
  }
  __syncthreads();
#pragma unroll
  for (int k = 0; k < 4; ++k) {
    const int j = k * 8 + w;
    const int n = n0 + j;
    if (n < N) {
      const int i = n * LANES + lane;
      Qt[i]  = tile[lane * 33 + j];
      T[i]   = 0.0f;
      lap[i] = 0.0f;
    }
  }
}

// ---------------------------------------------------------------------------
// SpMV scatter: one wave per COO entry, lane = batch. Coalesced 128B gather of
// T[col][*]; one-cacheline hardware float atomics into lap[row][*].
// ---------------------------------------------------------------------------
__global__ __launch_bounds__(256) void fem_spmv_kernel(
    const float* __restrict__ T, float* __restrict__ lap,
    const int* __restrict__ rows, const int* __restrict__ cols,
    const float* __restrict__ vals, int nnz) {
  const int e    = blockIdx.x * 8 + (threadIdx.x >> 5);
  const int lane = threadIdx.x & 31;
  if (e >= nnz) return;
  const int   c = cols[e];
  const int   r = rows[e];
  const float v = vals[e];
  const float g = T[c * LANES + lane] * v;
  float* p = &lap[r * LANES + lane];
  // Guaranteed hardware atomic (no-return form, tracked by STOREcnt).
  asm volatile("global_atomic_add_f32 %0, %1, off"
               :: "v"(p), "v"(g)
               : "memory");
}

// ---------------------------------------------------------------------------
// Update (scratch path, v4f): T += DT*(Q/rho + alpha*lap); re-zero lap for
// the next step; stream the plane to Ts[t][n][b] with non-temporal b128 stores
// so the history never evicts the L2-resident iteration state.
// ---------------------------------------------------------------------------
__global__ __launch_bounds__(256) void fem_update_kernel(
    const v4f* __restrict__ Qt, v4f* __restrict__ T,
    v4f* __restrict__ lap, v4f* __restrict__ Tsplane,
    const float* __restrict__ alpha_p, const float* __restrict__ rho_p, int BN4) {
  const int i = blockIdx.x * blockDim.x + threadIdx.x;
  if (i >= BN4) return;
  const float a   = *alpha_p;
  const float rho = *rho_p;
  const v4f l = lap[i];
  lap[i] = (v4f){0.0f, 0.0f, 0.0f, 0.0f};
  const v4f q = Qt[i];
  v4f t = T[i];
  t.x += DT_CONST * (q.x / rho + a * l.x);
  t.y += DT_CONST * (q.y / rho + a * l.y);
  t.z += DT_CONST * (q.z / rho + a * l.z);
  t.w += DT_CONST * (q.w / rho + a * l.w);
  T[i] = t;
  __builtin_nontemporal_store(t, &Tsplane[i]);
}

// Fallback (no scratch): write out[b][n][t] directly (strided).
__global__ __launch_bounds__(256) void fem_update_direct_kernel(
    const float* __restrict__ Qt, float* __restrict__ T,
    float* __restrict__ lap, float* __restrict__ out,
    const float* __restrict__ alpha_p, const float* __restrict__ rho_p,
    int N, int TS, int t) {
  const int i = blockIdx.x * blockDim.x + threadIdx.x;
  if (i >= N * LANES) return;
  const float a   = *alpha_p;
  const float rho = *rho_p;
  const float l   = lap[i];
  lap[i] = 0.0f;
  const float tn = T[i] + DT_CONST * (Qt[i] / rho + a * l);
  T[i] = tn;
  const int b = i & 31;
  const int n = i >> 5;
  __builtin_nontemporal_store(tn, &out[((size_t)b * N + n) * TS + t]);
}

// ---------------------------------------------------------------------------
// Emit: out[b][n][t] = Ts[t][n][b]. Tile = 32 n-values x 32 b x TST t-planes.
// Phase 1 pulls TST strided 4KB rows into LDS with CDNA5 async global->LDS
// B128 copies (ASYNCcnt, no VGPR round-trip), then s_wait_asynccnt + barrier.
// Phase 2 writes fully coalesced contiguous runs of out. TST templated so the
// index math uses constant-divisor mul-hi instead of runtime integer division.
// ---------------------------------------------------------------------------
template <int TST>
__global__ __launch_bounds__(256) void fem_emit_kernel(
    const float* __restrict__ Ts, float* __restrict__ out,
    int N, int BN) {
  extern __shared__ float smem[];
  const int n0 = blockIdx.x * 32;
  // Low 32 bits of a generic LDS pointer == LDS byte offset on gfx1250.
  const uint32_t lds_base = (uint32_t)(uintptr_t)(&smem[0]);

  if (n0 + 32 <= N) {
    // Fast path: async copy TST rows of 1024 floats (256 x b128 per row).
#pragma unroll
    for (int t = 0; t < TST; ++t) {
      const uint32_t voff = (uint32_t)(((size_t)t * BN + (size_t)n0 * LANES) *
                                       sizeof(float)) + threadIdx.x * 16u;
      const uint32_t loff = lds_base + (uint32_t)(t * PLANE + threadIdx.x * 4) * 4u;
      asm volatile("global_load_async_to_lds_b128 %0, %1, %2"
                   :: "v"(loff), "v"(voff), "s"(Ts)
                   : "memory");
    }
    asm volatile("s_wait_asynccnt 0" ::: "memory");
  } else {
    // Tail tile: guarded scalar loads.
    for (int idx = threadIdx.x; idx < TST * 1024; idx += 256) {
      const int t = idx >> 10;
      const int l = idx & 1023;
      const int j = l >> 5;
      const int b = l & 31;
      const int n = n0 + j;
      smem[t * PLANE + l] = (n < N) ? Ts[(size_t)t * BN + (size_t)n * LANES + b]
                                    : 0.0f;
    }
  }
  __syncthreads();

  const int total = LANES * 32 * TST;  // 32 b * 32 n * TST t
  for (int idx = threadIdx.x; idx < total; idx += 256) {
    const int b = idx / (32 * TST);        // constant divisor -> mul-hi
    const int r = idx % (32 * TST);
    const int j = r / TST;
    const int t = r % TST;
    const int n = n0 + j;
    if (n < N) {
      __builtin_nontemporal_store(smem[t * PLANE + j * 32 + b],
                                  &out[((size_t)b * N + n) * (size_t)TST + t]);
    }
  }
}

// Generic-TS variant (runtime divisor) for robustness.
__global__ __launch_bounds__(256) void fem_emit_kernel_dyn(
    const float* __restrict__ Ts, float* __restrict__ out,
    int N, int TS, int BN) {
  extern __shared__ float smem[];
  const int n0 = blockIdx.x * 32;
  for (int idx = threadIdx.x; idx < TS * 1024; idx += 256) {
    const int t = idx >> 10;
    const int l = idx & 1023;
    const int j = l >> 5;
    const int b = l & 31;
    const int n = n0 + j;
    smem[t * PLANE + l] = (n < N) ? Ts[(size_t)t * BN + (size_t)n * LANES + b]
                                  : 0.0f;
  }
  __syncthreads();
  const int total = LANES * 32 * TS;
  for (int idx = threadIdx.x; idx < total; idx += 256) {
    const int b = idx / (32 * TS);
    const int r = idx % (32 * TS);
    const int j = r / TS;
    const int t = r % TS;
    const int n = n0 + j;
    if (n < N) {
      __builtin_nontemporal_store(smem[t * PLANE + j * 32 + b],
                                  &out[((size_t)b * N + n) * (size_t)TS + t]);
    }
  }
}

// ---------------------------------------------------------------------------
// Host launcher
// ---------------------------------------------------------------------------
extern "C" void kernel_launch(void* const* d_in, const int* in_sizes, int n_in,
                              void* d_out, int out_size, void* d_ws, size_t ws_size,
                              hipStream_t stream) {
  const float* x     = (const float*)d_in[0];
  const float* alpha = (const float*)d_in[1];
  const float* rho_c = (const float*)d_in[2];
  const int*   rows  = (const int*)d_in[3];
  const int*   cols  = (const int*)d_in[4];
  const float* vals  = (const float*)d_in[5];
  float*       out   = (float*)d_out;

  const int BN  = in_sizes[0];       // B*N (B == 32 == wave32)
  const int N   = BN / LANES;
  const int nnz = in_sizes[3];
  const int TS  = out_size / BN;     // 13

  float* Qt  = (float*)d_ws;
  float* T   = Qt + BN;
  float* lap = T + BN;
  float* Ts  = lap + BN;
  const size_t need_full = (size_t)(3 + TS) * (size_t)BN * sizeof(float);
  const bool full = (ws_size >= need_full);

  const int tileBlocks = (N + 31) / 32;
  const int spmvBlocks = (nnz + 7) / 8;
  const int BN4        = BN / 4;                 // BN = 32*N -> always /4
  const int updBlocks  = (BN4 + 255) / 256;
  const int updBlocksS = (BN + 255) / 256;

  fem_init_kernel<<<tileBlocks, 256, 0, stream>>>(x, Qt, T, lap, N);

  for (int t = 0; t < TS; ++t) {
    fem_spmv_kernel<<<spmvBlocks, 256, 0, stream>>>(T, lap, rows, cols, vals, nnz);
    if (full) {
      fem_update_kernel<<<updBlocks, 256, 0, stream>>>(
          (const v4f*)Qt, (v4f*)T, (v4f*)lap,
          (v4f*)(Ts + (size_t)t * BN), alpha, rho_c, BN4);
    } else {
      fem_update_direct_kernel<<<updBlocksS, 256, 0, stream>>>(
          Qt, T, lap, out, alpha, rho_c, N, TS, t);
    }
  }

  if (full) {
    const size_t smem_bytes = (size_t)TS * PLANE * sizeof(float);
    if (TS == 13) {
      fem_emit_kernel<13><<<tileBlocks, 256, smem_bytes, stream>>>(Ts, out, N, BN);
    } else {
      fem_emit_kernel_dyn<<<tileBlocks, 256, smem_bytes, stream>>>(Ts, out, N, TS, BN);
    }
  }
}